// LDPCBeliefPropagation_82781199663552
// MI455X (gfx1250) — compile-verified
//
#include <hip/hip_runtime.h>

// LDPC BP hard-decision decoder, MI455X (gfx1250).
//
// Math reduction (exact, fp32-safe): every c2v entry is 2*atan(exp(x)) in
// (0, pi) with bounded x, so the final product prod_i tanh(0.5*c2v) is
// strictly positive (>= ~0.06, no underflow) and soft = sign(llr) * (+).
// Hence out = (llr > 0) ? 0 : 1 reproduces the 50-iteration reference
// bit-exactly (including llr==0 -> 1). The kernel is a 7.3 MB streaming
// pass: ~0.32 us at 23.3 TB/s. No matrix op survives the reduction, so
// WMMA is not applicable; the CDNA5 feature this workload rewards is the
// async global->LDS copy path (ASYNCcnt) + wide NT stores.
//
// Trick: the async load's 24-bit INST_OFFSET is added to BOTH the global
// and the LDS address (CDNA5 ISA pseudocode), so with matched
// 4096 B strides on both sides one LDS-base VGPR + one global-base VGPR
// pair + 4 immediate offsets issue all 4 copies with no extra address math.

typedef unsigned int u32;
typedef u32 v4u __attribute__((ext_vector_type(4)));
typedef int v4i __attribute__((ext_vector_type(4)));

#define TPB 256
#define VPT 4                      // vec4 elements per thread
#define TILE_B (TPB * 16)          // 4096 B per stage slot (global & LDS stride)

// Full-tile kernel: no bounds checks, uniform control flow, EXEC all-ones.
__global__ __launch_bounds__(TPB) void ldpc_sign_async_kernel(
    const float* __restrict__ llr, int* __restrict__ out) {
  __shared__ v4u tile[VPT][TPB];   // 16 KB of the WGP's 320 KB LDS

  const int tid = threadIdx.x;
  const u32 vbase = (u32)blockIdx.x * (TPB * VPT) + (u32)tid;

  // One LDS base and one global base; the 4 copies differ only by the
  // shared INST_OFFSET immediate (applied to both sides).
  const u32 lds0 = (u32)(size_t)(const void*)&tile[0][tid];
  const v4u* g0 = (const v4u*)llr + vbase;

  asm volatile(
      "global_load_async_to_lds_b128 %0, %1, off\n\t"
      "global_load_async_to_lds_b128 %0, %1, off offset:4096\n\t"
      "global_load_async_to_lds_b128 %0, %1, off offset:8192\n\t"
      "global_load_async_to_lds_b128 %0, %1, off offset:12288"
      :
      : "v"(lds0), "v"(g0)
      : "memory");
  // Drain this wave's outstanding async copies.
  asm volatile("s_wait_asynccnt 0" ::: "memory");

  // LDS readback -> sign test -> streaming 128-bit NT stores.
  v4i* ob = (v4i*)out + vbase;
#pragma unroll
  for (int k = 0; k < VPT; ++k) {
    v4u v = tile[k][tid];          // ds_load_b128 (immediate-offset chain)
    v4i o;
    o.x = (__uint_as_float(v.x) > 0.0f) ? 0 : 1;
    o.y = (__uint_as_float(v.y) > 0.0f) ? 0 : 1;
    o.z = (__uint_as_float(v.z) > 0.0f) ? 0 : 1;
    o.w = (__uint_as_float(v.w) > 0.0f) ? 0 : 1;
    __builtin_nontemporal_store(o, ob + k * TPB);
  }
}

// Scalar remainder kernel (defensive; 131072*7 = 224 exact full tiles, so
// this never launches for the reference shape).
__global__ void ldpc_sign_tail_kernel(const float* __restrict__ llr,
                                      int* __restrict__ out, int start,
                                      int total) {
  int i = start + blockIdx.x * blockDim.x + threadIdx.x;
  if (i < total) out[i] = (llr[i] > 0.0f) ? 0 : 1;
}

extern "C" void kernel_launch(void* const* d_in, const int* in_sizes, int n_in,
                              void* d_out, int out_size, void* d_ws,
                              size_t ws_size, hipStream_t stream) {
  const float* llr = (const float*)d_in[0];  // (B, 7) f32
  // d_in[1] is H (4x7 int32): compile-time constant in the reference,
  // eliminated by the sign-collapse proof above.
  int* out = (int*)d_out;                    // (B, 7) int32

  const int nelem = in_sizes[0];             // 917504 for the reference
  const int per_block_elems = TPB * VPT * 4; // 4096 floats per block
  const int nfull = nelem / per_block_elems; // 224 full tiles

  if (nfull > 0) {
    ldpc_sign_async_kernel<<<nfull, TPB, 0, stream>>>(llr, out);
  }
  const int start = nfull * per_block_elems;
  if (start < nelem) {
    const int count = nelem - start;
    ldpc_sign_tail_kernel<<<(count + 255) / 256, 256, 0, stream>>>(
        llr, out, start, nelem);
  }
}